// adaptive_implicit_trans_8091718386403
// MI455X (gfx1250) — compile-verified
//
#include <hip/hip_runtime.h>
#include <math.h>

typedef __attribute__((ext_vector_type(2))) float v2f;
typedef __attribute__((ext_vector_type(8))) float v8f;

#define HWPIX 65536   // 256*256
#define NCH   64
#define NSMP  16

// ---------------------------------------------------------------------------
// Kernel 1: build the 64x64 DCT basis K[o, c] into workspace.
// K[u*8+v, i*8+j] = c(u)c(v) cos((2i+1)u pi/16) cos((2j+1)v pi/16)
// ---------------------------------------------------------------------------
__global__ void build_dct_basis(float* __restrict__ kd) {
  int t = blockIdx.x * blockDim.x + threadIdx.x;
  if (t >= 64 * 64) return;
  int o = t >> 6, cc = t & 63;
  int u = o >> 3, v = o & 7;
  int i = cc >> 3, j = cc & 7;
  const float PI16 = 3.14159265358979323846f / 16.0f;
  const float r1 = 0.35355339059327373f;  // sqrt(1/8)
  const float r2 = 0.5f;                  // sqrt(2/8)
  float ru = (u == 0) ? r1 : r2;
  float rv = (v == 0) ? r1 : r2;
  kd[t] = ru * rv * cosf((float)((2 * i + 1) * u) * PI16)
                  * cosf((float)((2 * j + 1) * v) * PI16);
}

// ---------------------------------------------------------------------------
// Kernel 2: per-pixel 64x64 transform via V_WMMA_F32_16X16X4_F32.
// One wave -> 16 output channels x 16 pixels; K-dim (64 input channels)
// consumed in 16 steps of 4. Wave32 layouts per CDNA5 ISA 7.12.2:
//   A 16x4 f32 : lanes 0-15 K={0,1}, lanes 16-31 K={2,3}  (2 VGPRs)
//   B 4x16 f32 : VGPR v, lane-half h -> row K = v + 2h, col = lane%16
//   C/D 16x16  : VGPR r, lane-half h -> M = r + 8h, N = lane%16
// ---------------------------------------------------------------------------
__global__ __launch_bounds__(256) void dct_apply_wmma(
    const float* __restrict__ x,         // [16, 64, 256, 256]
    const float* __restrict__ backbone,  // [16, 64]
    const float* __restrict__ itw,       // [64]
    const float* __restrict__ kd,        // [64, 64] DCT basis
    float* __restrict__ out) {           // [16, 64, 256, 256]
  const int lane = threadIdx.x & 31;
  const int wid  = (int)((blockIdx.x * blockDim.x + threadIdx.x) >> 5);

  const int pb = wid & 4095;          // pixel block (16 pixels)
  const int mq = (wid >> 12) & 3;     // M quarter (16 output channels)
  const int n  = wid >> 14;           // sample
  const int h   = lane >> 4;          // lane half
  const int l16 = lane & 15;
  const int p   = pb * 16 + l16;      // this lane's pixel (B/C/D column)

  // Preload all A slices: lane's A row is m = mq*16 + l16; per step s the two
  // K columns (4s+2h, 4s+2h+1) are adjacent -> single b64 load each.
  const int m = mq * 16 + l16;
  v2f a[16];
#pragma unroll
  for (int s = 0; s < 16; ++s)
    a[s] = *(const v2f*)(kd + m * 64 + 4 * s + 2 * h);

  const float* xb = x + (size_t)n * NCH * HWPIX;
  v8f acc = {};
#pragma unroll
  for (int s = 0; s < 16; ++s) {
    const int c0 = 4 * s + 2 * h;    // this lane-half's B row pair
    v2f b;
    b.x = xb[(size_t)c0 * HWPIX + p];
    b.y = xb[(size_t)(c0 + 1) * HWPIX + p];
    // 8 args: (neg_a, A, neg_b, B, c_mod, C, reuse_a, reuse_b)
    acc = __builtin_amdgcn_wmma_f32_16x16x4_f32(
        false, a[s], false, b, (short)0, acc, false, false);
  }

  // Epilogue: per-row scale (backbone * it_weights) and store.
  float* ob = out + (size_t)n * NCH * HWPIX;
#pragma unroll
  for (int r = 0; r < 8; ++r) {
    const int mr = mq * 16 + r + 8 * h;      // C/D row for this VGPR/half
    const float sc = backbone[n * NCH + mr] * itw[mr];
    ob[(size_t)mr * HWPIX + p] = acc[r] * sc;
  }
}

// ---------------------------------------------------------------------------
extern "C" void kernel_launch(void* const* d_in, const int* in_sizes, int n_in,
                              void* d_out, int out_size, void* d_ws, size_t ws_size,
                              hipStream_t stream) {
  const float* x        = (const float*)d_in[0];  // [16,64,256,256] f32
  const float* backbone = (const float*)d_in[1];  // [16,64] f32
  const float* itw      = (const float*)d_in[2];  // [64] f32
  float* kd  = (float*)d_ws;                      // 4096 floats (16 KB)
  float* out = (float*)d_out;

  build_dct_basis<<<16, 256, 0, stream>>>(kd);

  // 16 samples * 4 M-quarters * 4096 pixel-blocks = 262144 waves
  // * 32 lanes / 256 threads per block = 32768 blocks
  dct_apply_wmma<<<32768, 256, 0, stream>>>(x, backbone, itw, kd, out);
}